// KGModel_4827543240965
// MI455X (gfx1250) — compile-verified
//
#include <hip/hip_runtime.h>

#define DD 128          // embedding dim
#define EE 32           // entities per item
#define TI 16           // items per block (WMMA M)
#define NRELR 25        // NUM_RELATIONS + 1 rows
#define ENT_MASK 60000  // padding entity index
#define ALPHA_ 0.2f
#define MASKV (-9000000000000000.0f)

typedef __attribute__((ext_vector_type(2))) float v2f;
typedef __attribute__((ext_vector_type(8))) float v8f;

__global__ void __launch_bounds__(256, 2)
kg_gat_fused(const float* __restrict__ it0, const float* __restrict__ it1,
             const float* __restrict__ en0, const float* __restrict__ en1,
             const float* __restrict__ re0, const float* __restrict__ re1,
             const float* __restrict__ fcw, const float* __restrict__ fcb,
             const float* __restrict__ ow,  const float* __restrict__ ob,
             const int*   __restrict__ ents,const int*   __restrict__ rels,
             float* __restrict__ out)
{
    __shared__ __align__(16) float s_item[2][TI][DD];   // 16 KB
    __shared__ __align__(16) float s_att[2][TI][EE];    // 4 KB (scores -> attention)
    __shared__ __align__(16) float s_A[TI][DD + 4];     // 8.25 KB WMMA A tile (h+item)
    __shared__ __align__(16) float s_fw[3 * DD];        // fc_w
    __shared__ __align__(16) float s_ob[DD];            // out_b
    __shared__ float s_reldot[2][32];                   // dot(rel_row, w_r)
    __shared__ float s_idot[2][TI];                     // dot(item,   w_h)

    const int tid = threadIdx.x;
    const int ib  = blockIdx.x * TI;

    // ---- stage constants + item embeddings ----
    for (int i = tid; i < 3 * DD; i += 256) s_fw[i] = fcw[i];
    if (tid < DD) s_ob[tid] = ob[tid];
    for (int i = tid; i < TI * DD; i += 256) {
        int m = i >> 7, d = i & (DD - 1);
        s_item[0][m][d] = it0[(size_t)(ib + m) * DD + d];
        s_item[1][m][d] = it1[(size_t)(ib + m) * DD + d];
    }
    __syncthreads();

    // ---- precompute per-relation dots and per-item dots ----
    if (tid < 64) {
        int c = tid >> 5, r = tid & 31;
        float s = 0.f;
        if (r < NRELR) {
            const float* rp = (c ? re1 : re0) + (size_t)r * DD;
            for (int d = 0; d < DD; ++d) s += rp[d] * s_fw[DD + d];
        }
        s_reldot[c][r] = s;
    } else if (tid < 96) {
        int t = tid - 64;
        int c = t >> 4, m = t & 15;
        float s = 0.f;
        for (int d = 0; d < DD; ++d) s += s_item[c][m][d] * s_fw[d];
        s_idot[c][m] = s;
    }
    __syncthreads();

    // ---- attention logits: thread -> (item = tid/16, slots {l, l+16}) ----
    const float bias = fcb[0];
    {
        int i = tid >> 4, l = tid & 15;
        for (int c = 0; c < 2; ++c) {
            const float* eb = c ? en1 : en0;
            for (int ss = 0; ss < 2; ++ss) {
                int s   = l + ss * 16;
                int idx = ents[(size_t)(ib + i) * EE + s];
                int rdx = rels[(size_t)(ib + i) * EE + s];
                const float4* ep = (const float4*)(eb + (size_t)idx * DD);
                const float4* wp = (const float4*)(s_fw + 2 * DD);
                float dot = 0.f;
                #pragma unroll 8
                for (int k = 0; k < DD / 4; ++k) {
                    float4 v = ep[k], w = wp[k];
                    dot += v.x * w.x + v.y * w.y + v.z * w.z + v.w * w.w;
                }
                float sc = s_idot[c][i] + s_reldot[c][rdx] + dot + bias;
                sc = sc > 0.f ? sc : ALPHA_ * sc;           // leaky relu
                if (idx == ENT_MASK) sc = MASKV;            // mask
                s_att[c][i][s] = sc;
            }
        }
    }
    __syncthreads();

    // ---- softmax over the 32 slots (one thread per (channel,item)) ----
    if (tid < 32) {
        int c = tid >> 4, i = tid & 15;
        float mx = -3.4e38f;
        for (int e = 0; e < EE; ++e) mx = fmaxf(mx, s_att[c][i][e]);
        float sum = 0.f;
        for (int e = 0; e < EE; ++e) {
            float v = __expf(s_att[c][i][e] - mx);
            s_att[c][i][e] = v; sum += v;
        }
        float inv = 1.f / sum;
        for (int e = 0; e < EE; ++e) s_att[c][i][e] *= inv;
    }
    __syncthreads();

    // ---- WMMA lane mapping (wave32): wave wv owns N-cols [16*wv, 16*wv+16) ----
    const int lane = tid & 31;
    const int wv   = tid >> 5;
    const int half = lane >> 4;      // K sub-half for A/B operands
    const int mrow = lane & 15;      // A row / C column index
    const int n0   = wv * 16 + mrow; // global output column

    float res[8];
    #pragma unroll
    for (int v = 0; v < 8; ++v) res[v] = 0.f;

    for (int c = 0; c < 2; ++c) {
        // ---- weighted entity aggregation: A = h + item (thread -> item x 8 dims) ----
        {
            int i = tid >> 4, dch = tid & 15;
            int d0 = dch * 8;
            const float* eb = c ? en1 : en0;
            float h[8];
            #pragma unroll
            for (int j = 0; j < 8; ++j) h[j] = 0.f;
            for (int e = 0; e < EE; ++e) {
                int   idx = ents[(size_t)(ib + i) * EE + e];
                float a   = s_att[c][i][e];
                const float4* p = (const float4*)(eb + (size_t)idx * DD + d0);
                float4 v0 = p[0], v1 = p[1];
                h[0] += a * v0.x; h[1] += a * v0.y; h[2] += a * v0.z; h[3] += a * v0.w;
                h[4] += a * v1.x; h[5] += a * v1.y; h[6] += a * v1.z; h[7] += a * v1.w;
            }
            #pragma unroll
            for (int j = 0; j < 8; ++j)
                s_A[i][d0 + j] = h[j] + s_item[c][i][d0 + j];
        }
        __syncthreads();

        // ---- (16x128) @ (128x128) via V_WMMA_F32_16X16X4_F32, 32 K-steps ----
        // A layout: lanes 0-15 M=0..15 {K, K+1}; lanes 16-31 M=0..15 {K+2, K+3}
        // B layout: lanes 0-15 N=n0    {K, K+1}; lanes 16-31 N=n0    {K+2, K+3}
        v8f acc = {};
        #pragma unroll 4
        for (int kk = 0; kk < DD / 4; ++kk) {
            int k = 4 * kk + 2 * half;
            v2f a; a.x = s_A[mrow][k];
                   a.y = s_A[mrow][k + 1];
            v2f b; b.x = ow[(size_t)k * DD + n0];
                   b.y = ow[(size_t)(k + 1) * DD + n0];
            acc = __builtin_amdgcn_wmma_f32_16x16x4_f32(
                false, a, false, b, (short)0, acc, false, false);
        }
        float bn = s_ob[n0];
        #pragma unroll
        for (int v = 0; v < 8; ++v) res[v] += fmaxf(acc[v] + bn, 0.f);
        __syncthreads();   // s_A reused by next channel
    }

    // ---- D layout: lanes 0-15 -> M = v, lanes 16-31 -> M = v + 8 ----
    #pragma unroll
    for (int v = 0; v < 8; ++v) {
        int m = v + 8 * half;
        out[(size_t)(ib + m) * DD + n0] = res[v] * 0.5f;
    }
}

extern "C" void kernel_launch(void* const* d_in, const int* in_sizes, int n_in,
                              void* d_out, int out_size, void* d_ws, size_t ws_size,
                              hipStream_t stream)
{
    const float* it0 = (const float*)d_in[0];
    const float* it1 = (const float*)d_in[1];
    const float* en0 = (const float*)d_in[2];
    const float* en1 = (const float*)d_in[3];
    const float* re0 = (const float*)d_in[4];
    const float* re1 = (const float*)d_in[5];
    const float* fcw = (const float*)d_in[6];
    const float* fcb = (const float*)d_in[7];
    const float* ow  = (const float*)d_in[8];
    const float* ob  = (const float*)d_in[9];
    const int*   ents= (const int*)d_in[10];
    const int*   rels= (const int*)d_in[11];
    float* out = (float*)d_out;

    int num_items = in_sizes[0] / DD;          // 20000
    int blocks = (num_items + TI - 1) / TI;    // 1250 (exact multiple of 16)

    kg_gat_fused<<<blocks, 256, 0, stream>>>(it0, it1, en0, en1, re0, re1,
                                             fcw, fcb, ow, ob, ents, rels, out);
}